// TorchLMHeadJSD_18245021073710
// MI455X (gfx1250) — compile-verified
//
#include <hip/hip_runtime.h>

// ---------------- problem constants (fixed by the reference) ----------------
constexpr int   kN    = 2048;   // tokens
constexpr int   kH    = 2048;   // hidden
constexpr int   kV    = 32000;  // vocab
constexpr float kBeta = 0.5f;

// ---------------- tiling ----------------
constexpr int ROWS = 64;   // token rows per workgroup
constexpr int COLS = 128;  // vocab cols per workgroup
constexpr int KT   = 32;   // K chunk staged in LDS
constexpr int LSTR = 36;   // padded LDS K stride: 16B-aligned b128 copies, conflict-free b64 frag reads

constexpr int A_TILE = ROWS * LSTR;        // floats per activation matrix tile
constexpr int B_TILE = COLS * LSTR;        // floats per weight matrix tile
constexpr int A_BUF  = 2 * A_TILE;         // student+teacher
constexpr int B_BUF  = 2 * B_TILE;
constexpr int SMEM_FLOATS = 2 * (A_BUF + B_BUF);             // double buffered
constexpr size_t SMEM_BYTES = SMEM_FLOATS * sizeof(float);   // 110592 B < 320 KB/WGP

typedef float v2f __attribute__((ext_vector_type(2)));
typedef float v8f __attribute__((ext_vector_type(8)));
typedef int   v4i __attribute__((ext_vector_type(4)));

// ---------------- CDNA5 async global->LDS copy (ASYNCcnt-tracked) ----------------
#if defined(__has_builtin)
#if __has_builtin(__builtin_amdgcn_global_load_async_to_lds_b128) && \
    __has_builtin(__builtin_amdgcn_s_wait_asynccnt)
#define USE_ASYNC_LDS 1
#endif
#endif

__device__ __forceinline__ void async_cp16(const float* gsrc, float* ldst) {
#if defined(USE_ASYNC_LDS)
  __builtin_amdgcn_global_load_async_to_lds_b128(
      (__attribute__((address_space(1))) v4i*)gsrc,
      (__attribute__((address_space(3))) v4i*)ldst, 0, 0);
#else
  *(float4*)ldst = *(const float4*)gsrc;
#endif
}

template <int IMM>
__device__ __forceinline__ void wait_async() {
#if defined(USE_ASYNC_LDS)
  __builtin_amdgcn_s_wait_asynccnt(IMM);
#endif
}

__device__ __forceinline__ float f_lo(unsigned long long v) {
  return __uint_as_float((unsigned)(v & 0xffffffffu));
}
__device__ __forceinline__ float f_hi(unsigned long long v) {
  return __uint_as_float((unsigned)(v >> 32));
}

// Merge (max, sumexp) pair packed into one 64-bit word with a CAS loop.
__device__ void merge_stat(unsigned long long* addr, float m, float z) {
  unsigned long long old = *(volatile unsigned long long*)addr;
  unsigned long long assumed;
  do {
    assumed = old;
    float M  = f_lo(assumed), Z = f_hi(assumed);
    float nM = fmaxf(M, m);
    float nZ = Z * __expf(M - nM) + z * __expf(m - nM);
    unsigned long long nv =
        ((unsigned long long)__float_as_uint(nZ) << 32) | __float_as_uint(nM);
    old = atomicCAS(addr, assumed, nv);
  } while (old != assumed);
}

__global__ void init_kernel(unsigned long long* stats, float* out) {
  int i = blockIdx.x * blockDim.x + threadIdx.x;
  if (i == 0) out[0] = 0.0f;
  if (i < 2 * kN) stats[i] = 0x00000000ff800000ull;  // max = -inf, sumexp = 0
}

// Issue the async copies for one K tile (12 b128 instructions per thread).
__device__ __forceinline__ void stage_tile(
    const float* __restrict__ s_in, const float* __restrict__ t_in,
    const float* __restrict__ Ws, const float* __restrict__ Wt,
    float* As, float* Bs, int n0, int v0, int kt, int tid) {
  const float* aSrc[2] = {s_in, t_in};
  const float* bSrc[2] = {Ws, Wt};
#pragma unroll
  for (int m = 0; m < 2; ++m) {
#pragma unroll
    for (int it = 0; it < 2; ++it) {          // A: 64x32 = 512 f4 / 256 thr
      int idx = tid + it * 256;
      int r = idx >> 3, kq = (idx & 7) << 2;
      async_cp16(aSrc[m] + (size_t)(n0 + r) * kH + kt + kq,
                 &As[m * A_TILE + r * LSTR + kq]);
    }
#pragma unroll
    for (int it = 0; it < 4; ++it) {          // B: 128x32 = 1024 f4 / 256 thr
      int idx = tid + it * 256;
      int r = idx >> 3, kq = (idx & 7) << 2;
      async_cp16(bSrc[m] + (size_t)(v0 + r) * kH + kt + kq,
                 &Bs[m * B_TILE + r * LSTR + kq]);
    }
  }
}

// PASS 1: GEMM + per-row (max, sumexp) for both student & teacher logits.
// PASS 2: GEMM recompute + fused JSD epilogue -> atomicAdd into scalar loss.
template <int PASS>
__global__ __launch_bounds__(256) void jsd_gemm(
    const float* __restrict__ s_in, const float* __restrict__ t_in,
    const float* __restrict__ Ws, const float* __restrict__ Wt,
    unsigned long long* __restrict__ stats, float* __restrict__ out) {
  extern __shared__ float smem[];
  float* Abase = smem;                 // [2 bufs][2 mats][ROWS*LSTR]
  float* Bbase = smem + 2 * A_BUF;     // [2 bufs][2 mats][COLS*LSTR]

  const int n0   = blockIdx.x * ROWS;
  const int v0   = blockIdx.y * COLS;
  const int tid  = threadIdx.x;
  const int lane = tid & 31;
  const int wid  = tid >> 5;       // 8 waves: 4 (row groups) x 2 (col groups)
  const int wrow = (wid & 3) * 16;
  const int wcol = (wid >> 2) * 64;
  const int l16  = lane & 15;
  const int hi   = lane >> 4;      // half-wave select (K / K+2 split per ISA layout)

  v8f acc[2][4];
  for (int m = 0; m < 2; ++m)
    for (int t = 0; t < 4; ++t)
      for (int e = 0; e < 8; ++e) acc[m][t][e] = 0.0f;

  // Pipeline prologue: stage tile 0 into buffer 0.
  stage_tile(s_in, t_in, Ws, Wt, Abase, Bbase, n0, v0, 0, tid);

  int cur = 0;
  for (int kt = 0; kt < kH; kt += KT) {
    const bool more = (kt + KT) < kH;
    const int nxt = cur ^ 1;
    if (more) {  // overlap next tile's HBM->LDS with this tile's WMMAs
      stage_tile(s_in, t_in, Ws, Wt, Abase + nxt * A_BUF, Bbase + nxt * B_BUF,
                 n0, v0, kt + KT, tid);
      wait_async<12>();   // loads retire in order: current buffer complete
    } else {
      wait_async<0>();
    }
    __syncthreads();      // all waves' copies for buf[cur] visible

    const float* As = Abase + cur * A_BUF;
    const float* Bs = Bbase + cur * B_BUF;
#pragma unroll
    for (int ks = 0; ks < KT; ks += 4) {
      const int ko = ks + 2 * hi;  // lanes 0-15: K,K+1  lanes 16-31: K+2,K+3
      v2f a0 = *(const v2f*)&As[0 * A_TILE + (wrow + l16) * LSTR + ko];
      v2f a1 = *(const v2f*)&As[1 * A_TILE + (wrow + l16) * LSTR + ko];
#pragma unroll
      for (int t = 0; t < 4; ++t) {
        v2f b0 = *(const v2f*)&Bs[0 * B_TILE + (wcol + t * 16 + l16) * LSTR + ko];
        acc[0][t] = __builtin_amdgcn_wmma_f32_16x16x4_f32(
            false, a0, false, b0, (short)0, acc[0][t], false, false);
        v2f b1 = *(const v2f*)&Bs[1 * B_TILE + (wcol + t * 16 + l16) * LSTR + ko];
        acc[1][t] = __builtin_amdgcn_wmma_f32_16x16x4_f32(
            false, a1, false, b1, (short)0, acc[1][t], false, false);
      }
    }
    __syncthreads();      // all waves done reading buf[cur]; safe to overwrite
    cur = nxt;
  }

  // C/D layout: VGPR vr holds (row = wrow + vr + 8*hi, col = wcol + t*16 + l16).
  if constexpr (PASS == 1) {
    for (int m = 0; m < 2; ++m) {
#pragma unroll
      for (int vr = 0; vr < 8; ++vr) {
        float mx = fmaxf(fmaxf(acc[m][0][vr], acc[m][1][vr]),
                         fmaxf(acc[m][2][vr], acc[m][3][vr]));
#pragma unroll
        for (int sh = 8; sh >= 1; sh >>= 1)
          mx = fmaxf(mx, __shfl_xor(mx, sh, 32));  // stays within 16-lane half
        float se = 0.0f;
#pragma unroll
        for (int t = 0; t < 4; ++t) se += __expf(acc[m][t][vr] - mx);
#pragma unroll
        for (int sh = 8; sh >= 1; sh >>= 1) se += __shfl_xor(se, sh, 32);
        if (l16 == 0) {
          int row = n0 + wrow + vr + 8 * hi;
          merge_stat(&stats[m * kN + row], mx, se);
        }
      }
    }
  } else {
    float lsum = 0.0f;
#pragma unroll
    for (int vr = 0; vr < 8; ++vr) {
      int row = n0 + wrow + vr + 8 * hi;
      unsigned long long sq = stats[row];       // student (q)
      unsigned long long sp = stats[kN + row];  // teacher (p)
      float Mq = f_lo(sq), lZq = __logf(f_hi(sq));
      float Mp = f_lo(sp), lZp = __logf(f_hi(sp));
#pragma unroll
      for (int t = 0; t < 4; ++t) {
        float lq = acc[0][t][vr] - Mq - lZq;
        float lp = acc[1][t][vr] - Mp - lZp;
        float q = __expf(lq), p = __expf(lp);
        float mm = p + kBeta * (q - p);
        float lm = __logf(fmaxf(mm, 1e-38f));
        lsum += kBeta * (p * (lp - lm)) + (1.0f - kBeta) * (q * (lq - lm));
      }
    }
#pragma unroll
    for (int sh = 16; sh >= 1; sh >>= 1) lsum += __shfl_xor(lsum, sh, 32);
    if (lane == 0) atomicAdd(out, lsum * (1.0f / kN));
  }
}

extern "C" void kernel_launch(void* const* d_in, const int* in_sizes, int n_in,
                              void* d_out, int out_size, void* d_ws,
                              size_t ws_size, hipStream_t stream) {
  (void)in_sizes; (void)n_in; (void)out_size; (void)ws_size;
  const float* s_in = (const float*)d_in[0];
  const float* t_in = (const float*)d_in[1];
  const float* Ws   = (const float*)d_in[2];
  const float* Wt   = (const float*)d_in[3];
  float* out = (float*)d_out;
  unsigned long long* stats = (unsigned long long*)d_ws;  // 2*kN*8 = 32 KB scratch

  init_kernel<<<(2 * kN + 255) / 256, 256, 0, stream>>>(stats, out);
  dim3 grid(kN / ROWS, kV / COLS);  // 32 x 250 workgroups
  jsd_gemm<1><<<grid, 256, SMEM_BYTES, stream>>>(s_in, t_in, Ws, Wt, stats, out);
  jsd_gemm<2><<<grid, 256, SMEM_BYTES, stream>>>(s_in, t_in, Ws, Wt, stats, out);
}